// Decoder_8289286881955
// MI455X (gfx1250) — compile-verified
//
#include <hip/hip_runtime.h>
#include <stdint.h>

// ---------------- problem constants (from reference) ----------------
constexpr int kB      = 8;        // samples
constexpr int kG      = 32;       // GRID/STRIDE blocks per axis
constexpr int kBmaxN  = kB * kG * kG * kG;   // 262144 segments
constexpr int kBins   = 256;      // 8-bit radix digits
constexpr int kThreads = 256;
constexpr int kTileDW  = 1024;    // dwords per async tile (4KB)

// ---------------- workspace layout (uint32 units) ----------------
// [0, 262144)              : per-segment max keys (flipped floats)
// [262144, 262144+2048)    : global histograms, 8 samples x 256 bins
// then prefix[8], kremain[8], thr[8]
// [524288, 524288+N)       : masked key array (32 MB)
#define WS_BMAX   0u
#define WS_HIST   ((uint32_t)kBmaxN)
#define WS_PREFIX (WS_HIST + (uint32_t)(kB * kBins))
#define WS_KREM   (WS_PREFIX + 8u)
#define WS_THR    (WS_KREM + 8u)
#define WS_KEY    (1u << 19)      // 524288 > WS_THR+8

// order-preserving float <-> uint mapping
__device__ __forceinline__ uint32_t flipf(float f) {
  uint32_t u = __float_as_uint(f);
  return (u & 0x80000000u) ? ~u : (u | 0x80000000u);
}
__device__ __forceinline__ float unflipf(uint32_t u) {
  uint32_t v = (u & 0x80000000u) ? (u & 0x7FFFFFFFu) : ~u;
  return __uint_as_float(v);
}

// ---------------- kernels ----------------

__global__ void init_ws(uint32_t* __restrict__ ws, const int* __restrict__ tnum, int M) {
  int i = blockIdx.x * blockDim.x + threadIdx.x;
  if (i < kBmaxN)      ws[WS_BMAX + i] = 0u;          // 0 < flip(any finite float)
  if (i < kB * kBins)  ws[WS_HIST + i] = 0u;
  if (i < kB) {
    ws[WS_PREFIX + i] = 0u;
    ws[WS_KREM   + i] = (uint32_t)(M - *tnum);        // k (1-indexed kth smallest)
    ws[WS_THR    + i] = 0u;
  }
}

__device__ __forceinline__ int seg_of(int4 c) {
  // c = (batch, x, y, z)
  return ((c.x * kG + (c.y >> 2)) * kG + (c.z >> 2)) * kG + (c.w >> 2);
}

__global__ void block_max(const float* __restrict__ pred, const int4* __restrict__ coords,
                          uint32_t* __restrict__ bmax, int N) {
  int i = blockIdx.x * blockDim.x + threadIdx.x;
  if (i >= N) return;
  int4 c = coords[i];
  atomicMax(&bmax[seg_of(c)], flipf(pred[i]));
}

__global__ void build_key(const float* __restrict__ pred, const int4* __restrict__ coords,
                          const uint32_t* __restrict__ bmax, uint32_t* __restrict__ key, int N) {
  int i = blockIdx.x * blockDim.x + threadIdx.x;
  if (i >= N) return;
  __builtin_prefetch(&coords[i + 8192], 0, 0);        // global_prefetch_b8
  int4 c = coords[i];
  uint32_t k = flipf(pred[i]);
  // local max (pred == block_max) -> +inf sentinel, exactly matching jnp.where(mask, v, inf)
  key[i] = (k == bmax[seg_of(c)]) ? 0xFFFFFFFFu : k;
}

// histogram pass with async global->LDS double-buffered streaming
__global__ __launch_bounds__(kThreads)
void hist_pass(const uint32_t* __restrict__ key, uint32_t* __restrict__ ws,
               int N, int M, int shift, int tiles_per_wg) {
  __shared__ uint32_t smem[2 * kTileDW + kB * kBins + kB];
  uint32_t* tiles = smem;                    // LDS offset 0 (sole LDS object)
  uint32_t* hist  = smem + 2 * kTileDW;
  uint32_t* pfx   = hist + kB * kBins;

  const int tid = threadIdx.x;
  for (int j = tid; j < kB * kBins; j += kThreads) hist[j] = 0u;
  if (tid < kB) pfx[tid] = ws[WS_PREFIX + tid];
  __syncthreads();

  const long long chunk_base = (long long)blockIdx.x * (long long)tiles_per_wg * kTileDW;
  long long rem = (long long)N - chunk_base;
  if (rem <= 0) return;                               // block-uniform
  int ntiles = (int)((rem + kTileDW - 1) / kTileDW);
  if (ntiles > tiles_per_wg) ntiles = tiles_per_wg;

  auto issue = [&](int t) {
    long long s0 = chunk_base + (long long)t * kTileDW + (long long)tid * 4;
    long long smax = (long long)N - 4;                // N % 4 == 0
    if (s0 > smax) s0 = smax;                         // OOB lanes clamp (slots skipped later)
    uint32_t lds_off = (uint32_t)((((t & 1) * kTileDW) + tid * 4) * 4); // bytes
    const uint32_t* g = key + s0;
    asm volatile("global_load_async_to_lds_b128 %0, %1, off"
                 :: "v"(lds_off), "v"(g) : "memory");
  };

  issue(0);
  for (int t = 0; t < ntiles; ++t) {
    if (t + 1 < ntiles) {
      issue(t + 1);
      asm volatile("s_wait_asynccnt 1" ::: "memory"); // tile t retired (in-order completion)
    } else {
      asm volatile("s_wait_asynccnt 0" ::: "memory");
    }
    __syncthreads();                                  // tile t visible block-wide
    const uint32_t* buf = tiles + (t & 1) * kTileDW;
    const long long base = chunk_base + (long long)t * kTileDW;
#pragma unroll
    for (int j = 0; j < 4; ++j) {
      int slot = tid * 4 + j;
      long long gi = base + slot;
      if (gi < N) {
        uint32_t k = buf[slot];
        uint32_t b = (uint32_t)(gi / M);
        bool ok = (shift == 24) || ((k >> (shift + 8)) == pfx[b]);
        if (ok) atomicAdd(&hist[b * kBins + ((k >> shift) & 0xFFu)], 1u);  // ds_add
      }
    }
    __syncthreads();                                  // protect buffer from next issue (WAR)
  }

  for (int j = tid; j < kB * kBins; j += kThreads) {
    uint32_t v = hist[j];
    if (v) atomicAdd(&ws[WS_HIST + j], v);
  }
}

// pick the digit containing the kth element; on last pass produce the float threshold
__global__ void scan_pass(uint32_t* __restrict__ ws, int last) {
  int b = threadIdx.x;
  if (b >= kB) return;
  uint32_t pfx = ws[WS_PREFIX + b];
  uint32_t kr  = ws[WS_KREM + b];
  uint32_t cum = 0, digit = 255u, krn = kr;
  bool found = false;
  for (int bin = 0; bin < kBins; ++bin) {
    uint32_t h = ws[WS_HIST + b * kBins + bin];
    ws[WS_HIST + b * kBins + bin] = 0u;               // re-zero for next pass
    if (!found && cum + h >= kr) { digit = (uint32_t)bin; krn = kr - cum; found = true; }
    cum += h;
  }
  uint32_t npfx = (pfx << 8) | digit;
  ws[WS_PREFIX + b] = npfx;
  ws[WS_KREM   + b] = krn;
  if (last) ws[WS_THR + b] = __float_as_uint(unflipf(npfx));
}

__global__ void apply_keep(const float* __restrict__ pred, const uint32_t* __restrict__ key,
                           const uint32_t* __restrict__ ws, float* __restrict__ out,
                           int N, int M) {
  int i = blockIdx.x * blockDim.x + threadIdx.x;
  if (i >= N) return;
  uint32_t b = (uint32_t)(i / M);
  float thr = __uint_as_float(ws[WS_THR + b]);
  float v = pred[i];
  bool keep = (v > thr) || (key[i] == 0xFFFFFFFFu);   // > kth value, or local max
  out[i] = keep ? v : 0.0f;
}

// ---------------- host launcher ----------------
extern "C" void kernel_launch(void* const* d_in, const int* in_sizes, int n_in,
                              void* d_out, int out_size, void* d_ws, size_t ws_size,
                              hipStream_t stream) {
  const float* pred   = (const float*)d_in[0];
  const int4*  coords = (const int4*)d_in[1];
  const int*   tnum   = (const int*)d_in[2];
  const int N = in_sizes[0];
  const int M = N / kB;
  uint32_t* ws  = (uint32_t*)d_ws;
  uint32_t* key = ws + WS_KEY;

  init_ws<<<(kBmaxN + 255) / 256, 256, 0, stream>>>(ws, tnum, M);

  const int blocks = (N + 255) / 256;
  block_max<<<blocks, 256, 0, stream>>>(pred, coords, ws + WS_BMAX, N);
  build_key<<<blocks, 256, 0, stream>>>(pred, coords, ws + WS_BMAX, key, N);

  const int NUM_WG = 512;
  long long total_tiles = ((long long)N + kTileDW - 1) / kTileDW;
  int tiles_per_wg = (int)((total_tiles + NUM_WG - 1) / NUM_WG);

  for (int p = 0; p < 4; ++p) {
    int shift = 24 - 8 * p;
    hist_pass<<<NUM_WG, kThreads, 0, stream>>>(key, ws, N, M, shift, tiles_per_wg);
    scan_pass<<<1, 32, 0, stream>>>(ws, p == 3);
  }

  apply_keep<<<blocks, 256, 0, stream>>>(pred, key, ws, (float*)d_out, N, M);
}